// Synchronizer_50337016709331
// MI455X (gfx1250) — compile-verified
//
#include <hip/hip_runtime.h>
#include <stdint.h>

#define D_MODEL 256
#define D2      65536
#define BATCH   512
#define THREADS 1024
#define F4_PER_ROW (D2 / 4)            // 16384 float4 per row
#define ITERS (F4_PER_ROW / THREADS)   // 16
#define LN_EPS 1e-5f

typedef float v4f __attribute__((ext_vector_type(4)));

__global__ __launch_bounds__(THREADS)
void synchronizer_kernel(const float* __restrict__ state,
                         const float* __restrict__ alpha_in,
                         const float* __restrict__ beta_in,
                         const float* __restrict__ w_in,
                         const float* __restrict__ gamma,
                         const float* __restrict__ lnbeta,
                         float* __restrict__ out)
{
    __shared__ float sState[D_MODEL];   // one state row
    __shared__ float sRed[64];          // 32 waves x {sum, sumsq}
    __shared__ float sBcast[2];         // mu, inv_std

    const int tid = threadIdx.x;
    const int row = blockIdx.x;
    const size_t base = (size_t)row * (size_t)D2;

    // --- CDNA5 async global->LDS copy of the state row (no VGPR round-trip) ---
    if (tid < D_MODEL) {
        uint32_t ldsAddr = (uint32_t)(uintptr_t)(&sState[tid]);
        uint64_t gAddr   = (uint64_t)(uintptr_t)(state + (size_t)row * D_MODEL + tid);
        asm volatile("global_load_async_to_lds_b32 %0, %1, off\n\t"
                     "s_wait_asynccnt 0"
                     :: "v"(ldsAddr), "v"(gAddr) : "memory");
    }
    __syncthreads();

    const v4f* a4 = (const v4f*)(alpha_in + base);
    const v4f* b4 = (const v4f*)(beta_in  + base);
    const v4f* w4 = (const v4f*)(w_in     + base);
    v4f* oS = (v4f*)(out + base);                                  // sync
    v4f* oA = (v4f*)(out + (size_t)BATCH * D2 + base);             // alpha'
    v4f* oB = (v4f*)(out + 2 * (size_t)BATCH * D2 + base);         // beta'

    // ---------------- Phase 1: elementwise update + row accumulation --------
    float sum = 0.f, sq = 0.f;
#pragma unroll 4
    for (int k = 0; k < ITERS; ++k) {
        const int f = tid + k * THREADS;           // float4 index in row
        // read-once inputs: non-temporal, keep L2 free for alpha'/beta'
        v4f wv = __builtin_nontemporal_load(w4 + f);
        v4f av = __builtin_nontemporal_load(a4 + f);
        v4f bv = __builtin_nontemporal_load(b4 + f);

        const float left = sState[f >> 6];                 // e / 256 (uniform per wave)
        const v4f   sr   = ((const v4f*)sState)[f & 63];   // e % 256 block of 4

        v4f an, bn;
#pragma unroll
        for (int j = 0; j < 4; ++j) {
            const float w2 = wv[j] * wv[j];
            const float p  = left * sr[j];
            const float a2 = __builtin_fmaf(w2, av[j], p);
            const float b2 = __builtin_fmaf(w2, bv[j], 1.0f);
            an[j] = a2; bn[j] = b2;
            const float s = a2 * rsqrtf(b2);
            sum += s;
            sq = __builtin_fmaf(s, s, sq);
        }
        oA[f] = an;   // default RT temporal hint: stays hot in L2 for phase 2
        oB[f] = bn;
    }

    // ---------------- Row reduction (wave32 shuffles + LDS across waves) ----
#pragma unroll
    for (int off = 16; off > 0; off >>= 1) {
        sum += __shfl_down(sum, off, 32);
        sq  += __shfl_down(sq,  off, 32);
    }
    const int lane = tid & 31;
    const int wid  = tid >> 5;
    if (lane == 0) { sRed[wid] = sum; sRed[32 + wid] = sq; }
    __syncthreads();
    if (tid < 32) {
        float s2 = sRed[tid];
        float q2 = sRed[32 + tid];
#pragma unroll
        for (int off = 16; off > 0; off >>= 1) {
            s2 += __shfl_down(s2, off, 32);
            q2 += __shfl_down(q2, off, 32);
        }
        if (tid == 0) {
            const float mu  = s2 * (1.0f / (float)D2);
            const float var = q2 * (1.0f / (float)D2) - mu * mu;
            sBcast[0] = mu;
            sBcast[1] = rsqrtf(var + LN_EPS);
        }
    }
    __syncthreads();
    const float mu  = sBcast[0];
    const float inv = sBcast[1];

    // ---------------- Phase 2: normalize (alpha'/beta' re-read hits L2) ----
    const v4f* g4  = (const v4f*)gamma;
    const v4f* be4 = (const v4f*)lnbeta;
#pragma unroll 4
    for (int k = 0; k < ITERS; ++k) {
        const int f = tid + k * THREADS;
        const v4f an = oA[f];   // same lane wrote these addresses: ordered, L2-hot
        const v4f bn = oB[f];
        const v4f g  = g4[f];
        const v4f be = be4[f];
        v4f o;
#pragma unroll
        for (int j = 0; j < 4; ++j) {
            const float s = an[j] * rsqrtf(bn[j]);
            o[j] = __builtin_fmaf((s - mu) * inv, g[j], be[j]);
        }
        __builtin_nontemporal_store(o, oS + f);   // write-once output
    }
}

extern "C" void kernel_launch(void* const* d_in, const int* in_sizes, int n_in,
                              void* d_out, int out_size, void* d_ws, size_t ws_size,
                              hipStream_t stream)
{
    (void)in_sizes; (void)n_in; (void)out_size; (void)d_ws; (void)ws_size;
    const float* state  = (const float*)d_in[0];
    const float* alpha  = (const float*)d_in[1];
    const float* beta   = (const float*)d_in[2];
    const float* w      = (const float*)d_in[3];
    const float* gamma  = (const float*)d_in[4];
    const float* lnbeta = (const float*)d_in[5];
    float* out = (float*)d_out;

    synchronizer_kernel<<<BATCH, THREADS, 0, stream>>>(
        state, alpha, beta, w, gamma, lnbeta, out);
}